// CandidateFinder_74552042324472
// MI455X (gfx1250) — compile-verified
//
#include <hip/hip_runtime.h>
#include <stdint.h>

#define BB 4
#define SS 4096
#define DD 64
#define HH 16
#define KMAX 64
#define NEGV (-1e30f)

typedef float v2f __attribute__((ext_vector_type(2)));
typedef float v8f __attribute__((ext_vector_type(8)));

// ---------------------------------------------------------------------------
// Kernel A: pack sign bits of each token row into a 64-bit mask.
// thread = token; tokens 0..B*S-1 -> query, B*S..2*B*S-1 -> key.
// ---------------------------------------------------------------------------
__global__ void masks_kernel(const float* __restrict__ q,
                             const float* __restrict__ k,
                             unsigned long long* __restrict__ qmask,
                             unsigned long long* __restrict__ kmask) {
  int t = blockIdx.x * blockDim.x + threadIdx.x;
  const int NT = BB * SS;
  const float* src = (t < NT) ? q : k;
  unsigned long long* dst = (t < NT) ? qmask : kmask;
  int tok = (t < NT) ? t : t - NT;
  const float4* row = (const float4*)(src + (size_t)tok * DD);
  unsigned long long m = 0ull;
#pragma unroll
  for (int j = 0; j < DD / 4; ++j) {
    float4 v = row[j];
    m |= ((unsigned long long)(v.x > 0.f)) << (4 * j + 0);
    m |= ((unsigned long long)(v.y > 0.f)) << (4 * j + 1);
    m |= ((unsigned long long)(v.z > 0.f)) << (4 * j + 2);
    m |= ((unsigned long long)(v.w > 0.f)) << (4 * j + 3);
  }
  dst[tok] = m;
}

// ---------------------------------------------------------------------------
// Kernel B: LSH hash via fp32 WMMA.  proj = X[16x64] @ W[64x16] per wave,
// chained V_WMMA_F32_16X16X4_F32 over K.  Then hash = (sum_n floor(proj/4)) & 63.
// A layout (ISA 7.12.2, 32-bit A 16x4): lanes 0-15 -> M=lane, K = {k0, k0+1};
// lanes 16-31 -> M=lane-16, K = {k0+2, k0+3}.  B 4x16 symmetric (row striped
// across lanes; lane half selects K pair).  C: VGPR r, lanes 0-15 M=r,
// lanes 16-31 M=8+r, N=lane&15.
// ---------------------------------------------------------------------------
__global__ void lsh_hash_kernel(const float* __restrict__ q,
                                const float* __restrict__ k,
                                const float* __restrict__ W,
                                unsigned char* __restrict__ qh,
                                unsigned char* __restrict__ kh) {
  const int NT = BB * SS;       // tokens per source
  const int TILES = NT / 16;    // 1024 tiles per source
  int wid = (blockIdx.x * blockDim.x + threadIdx.x) >> 5;
  int lane = threadIdx.x & 31;
  const float* src = (wid < TILES) ? q : k;
  unsigned char* dst = (wid < TILES) ? qh : kh;
  int t0 = ((wid < TILES) ? wid : wid - TILES) * 16;
  int half = lane >> 4;
  int m = lane & 15;

  v8f c = {};
#pragma unroll
  for (int kk = 0; kk < DD / 4; ++kk) {
    int k0 = kk * 4 + half * 2;
    const float* ar = src + (size_t)(t0 + m) * DD + k0;
    v2f a;
    a.x = ar[0];
    a.y = ar[1];
    v2f bmat;
    bmat.x = W[k0 * HH + m];
    bmat.y = W[(k0 + 1) * HH + m];
    // 8 args: (neg_a, A, neg_b, B, c_mod, C, reuse_a, reuse_b)
    c = __builtin_amdgcn_wmma_f32_16x16x4_f32(false, a, false, bmat,
                                              (short)0, c, false, false);
  }

  // codes = floor(proj / 4); reduce over N (16 lanes per half); hash = sum mod 64
#pragma unroll
  for (int r = 0; r < 8; ++r) {
    int s = (int)floorf(c[r] * 0.25f);
    s += __shfl_xor(s, 1, 32);
    s += __shfl_xor(s, 2, 32);
    s += __shfl_xor(s, 4, 32);
    s += __shfl_xor(s, 8, 32);
    if (m == r) dst[t0 + r + half * 8] = (unsigned char)(s & 63);
  }
}

// ---------------------------------------------------------------------------
// Kernel C: one wave per query.  Scan all 4096 keys; a key survives iff
//   hash equal  &&  batch-3 sign masks equal  &&  inserted (batch-0 6-bit prefix).
// Exact fp32 dot for survivors only; iterative wave-max top-64 with
// jax.lax.top_k tie-breaking (desc value, asc index); NEG entries -> -1.
// ---------------------------------------------------------------------------
__global__ void __launch_bounds__(64) match_topk_kernel(
    const float* __restrict__ q, const float* __restrict__ k,
    const unsigned char* __restrict__ qh, const unsigned char* __restrict__ kh,
    const unsigned long long* __restrict__ qmask,
    const unsigned long long* __restrict__ kmask,
    int* __restrict__ out) {
  __shared__ float lsim[2][SS];
  __shared__ float lq[2][DD];
  int w = threadIdx.x >> 5;
  int lane = threadIdx.x & 31;
  int t = blockIdx.x * 2 + w;   // query token 0..B*S-1
  int b = t >> 12;              // / S
  int qi = t & (SS - 1);

  // cache query row in LDS
  const float* qrow = q + (size_t)t * DD;
  lq[w][2 * lane]     = qrow[2 * lane];
  lq[w][2 * lane + 1] = qrow[2 * lane + 1];
  __syncthreads();

  int qhash = (int)qh[t];
  unsigned long long qm3 = qmask[3 * SS + qi];
  const unsigned char* khb = kh + (size_t)b * SS;

  int mcount = 0;
  for (int j = 0; j < SS / 32; ++j) {
    int ki = j * 32 + lane;
    bool flag = ((int)khb[ki] == qhash) &&
                (kmask[3 * SS + ki] == qm3) &&
                (((qmask[ki] ^ kmask[ki]) & 0x3Full) == 0ull);
    float sv = NEGV;
    if (flag) {
      const float4* kr = (const float4*)(k + (size_t)(b * SS + ki) * DD);
      float acc = 0.f;
#pragma unroll
      for (int d4 = 0; d4 < DD / 4; ++d4) {
        float4 v = kr[d4];
        acc = fmaf(lq[w][4 * d4 + 0], v.x, acc);
        acc = fmaf(lq[w][4 * d4 + 1], v.y, acc);
        acc = fmaf(lq[w][4 * d4 + 2], v.z, acc);
        acc = fmaf(lq[w][4 * d4 + 3], v.w, acc);
      }
      sv = acc;
    }
    lsim[w][ki] = sv;   // slot ki only ever touched by lane == ki & 31
    mcount += (int)flag;
  }
#pragma unroll
  for (int off = 16; off > 0; off >>= 1) mcount += __shfl_xor(mcount, off, 32);

  int* orow = out + (size_t)t * KMAX;
  if (mcount == 0) {  // overwhelmingly common: no candidates at all
    orow[lane] = -1;
    orow[lane + 32] = -1;
    return;
  }

  for (int it = 0; it < KMAX; ++it) {
    float bv = NEGV;
    int bi = SS;
    for (int j = 0; j < SS / 32; ++j) {
      int ki = j * 32 + lane;
      float v = lsim[w][ki];
      if (v > bv) { bv = v; bi = ki; }   // within-lane ties keep lower index
    }
#pragma unroll
    for (int off = 16; off > 0; off >>= 1) {
      float ov = __shfl_xor(bv, off, 32);
      int oi = __shfl_xor(bi, off, 32);
      if (ov > bv || (ov == bv && oi < bi)) { bv = ov; bi = oi; }
    }
    if (bv > NEGV) {
      if ((bi & 31) == lane) lsim[w][bi] = NEGV;  // owner lane clears its slot
      if (lane == 0) orow[it] = bi;
    } else {
      if (lane == 0) orow[it] = -1;
    }
  }
}

// ---------------------------------------------------------------------------
extern "C" void kernel_launch(void* const* d_in, const int* in_sizes, int n_in,
                              void* d_out, int out_size, void* d_ws, size_t ws_size,
                              hipStream_t stream) {
  const float* qup = (const float*)d_in[0];   // [B,S,D] f32
  const float* kup = (const float*)d_in[1];   // [B,S,D] f32
  const float* W   = (const float*)d_in[2];   // [D,H]   f32
  // d_in[3] = head_idx (unused by the reference computation)
  int* out = (int*)d_out;                     // [B,S,KMAX] int32

  char* ws = (char*)d_ws;
  unsigned long long* qmask = (unsigned long long*)(ws);            // 128 KB
  unsigned long long* kmask = (unsigned long long*)(ws + 131072);   // 128 KB
  unsigned char* qh = (unsigned char*)(ws + 262144);                //  16 KB
  unsigned char* kh = (unsigned char*)(ws + 262144 + 16384);        //  16 KB

  // A: 2*B*S tokens, one thread each
  masks_kernel<<<(2 * BB * SS) / 256, 256, 0, stream>>>(qup, kup, qmask, kmask);
  // B: 2*B*S/16 = 2048 waves -> 65536 threads
  lsh_hash_kernel<<<256, 256, 0, stream>>>(qup, kup, W, qh, kh);
  // C: one wave per query, 2 waves per block
  match_topk_kernel<<<(BB * SS) / 2, 64, 0, stream>>>(qup, kup, qh, kh, qmask,
                                                      kmask, out);
}